// SiameseTangentLayer_26560077758935
// MI455X (gfx1250) — compile-verified
//
#include <hip/hip_runtime.h>

// ltangent_distance, factored form:
//   dist[b,k] = ||y_k - x_b||^2 - 2||u||^2 + u^T G_k u,
//   u = v_k - (x_b @ Omega_k),  v_k = Omega_k^T y_k,  G_k = Omega_k^T Omega_k
// GEMMs via V_WMMA_F32_16X16X4_F32 (exact fp32, matches reference dtype).
// Global->LDS staging via GLOBAL_LOAD_ASYNC_TO_LDS_B128 (ASYNCcnt),
// double-buffered against the WMMA loop.

typedef __attribute__((ext_vector_type(2))) float v2f;
typedef __attribute__((ext_vector_type(8))) float v8f;

#define B_TOT   1024
#define K_PROTO 62
#define D_DIM   512
#define S_DIM   64
#define TILE_M  64
#define DCHUNK  128
#define NCHUNK  (D_DIM / DCHUNK)   // 4
#define XPAD    132   // 128 + 4 -> column reads hit distinct LDS banks
#define UPAD    65    // 64 + 1  -> column reads hit distinct LDS banks

// One async 16-byte global->LDS copy per lane (tracked by ASYNCcnt).
__device__ __forceinline__ void async_copy16(unsigned lds_off, const float* gsrc) {
    asm volatile("global_load_async_to_lds_b128 %0, %1, off"
                 :: "v"(lds_off), "v"((unsigned long long)(uintptr_t)gsrc)
                 : "memory");
}
#define WAIT_ASYNC(n) asm volatile("s_wait_asynccnt %0" :: "i"(n) : "memory")

// fp32 WMMA fragment mapping (ISA 7.12.2, 32-bit 16x4 A / 4x16 B / 16x16 C):
//   lmod = lane&15; lhalf = lane>>4; koff = 2*lhalf
//   A vgpr0 = A[m=lmod, kk+koff],  vgpr1 = A[m=lmod, kk+koff+1]
//   B vgpr0 = B[kk+koff, n=lmod],  vgpr1 = B[kk+koff+1, n=lmod]
//   C vgpr j = C[m = 16*mi + j + 8*lhalf, n = 16*ni + lmod]

// ---------------------------------------------------------------------------
// Kernel A: per prototype k, G_k = Omega^T Omega (64x64) and v_k = Omega^T y_k.
// ---------------------------------------------------------------------------
__global__ __launch_bounds__(512)
void proto_prep_kernel(const float* __restrict__ y,       // [K, D]
                       const float* __restrict__ omegas,  // [K, D, S]
                       float* __restrict__ G_ws,          // [K, 64, 64]
                       float* __restrict__ v_ws)          // [K, 64]
{
    const int k = blockIdx.x;
    const float* om = omegas + (size_t)k * D_DIM * S_DIM;

    __shared__ float lds_v[S_DIM];

    const int t     = threadIdx.x;
    const int lane  = t & 31;
    const int wave  = t >> 5;
    const int mi    = wave >> 2;
    const int ni    = wave & 3;
    const int lmod  = lane & 15;
    const int lhalf = lane >> 4;
    const int koff  = lhalf * 2;

    if (t < S_DIM) lds_v[t] = 0.f;
    __syncthreads();

    // G = Omega^T * Omega : M=N=64, reduction over D=512 (128 wmma steps)
    v8f acc = {};
    const int m = mi * 16 + lmod;
    const int n = ni * 16 + lmod;
    #pragma unroll 4
    for (int kk = 0; kk < D_DIM; kk += 4) {
        v2f a, b;
        a.x = om[(size_t)(kk + koff)     * S_DIM + m];  // Omega^T[m, d]
        a.y = om[(size_t)(kk + koff + 1) * S_DIM + m];
        b.x = om[(size_t)(kk + koff)     * S_DIM + n];  // Omega[d, n]
        b.y = om[(size_t)(kk + koff + 1) * S_DIM + n];
        acc = __builtin_amdgcn_wmma_f32_16x16x4_f32(false, a, false, b,
                                                    (short)0, acc, false, false);
    }
    float* Gk = G_ws + (size_t)k * S_DIM * S_DIM;
    #pragma unroll
    for (int j = 0; j < 8; ++j) {
        const int row = mi * 16 + j + lhalf * 8;
        Gk[row * S_DIM + n] = acc[j];
    }

    // v_k[s] = sum_d Omega[d,s] * y_k[d]  (8 partial chunks of 64, LDS atomics)
    {
        const int s     = t & 63;
        const int chunk = t >> 6;
        const float* yk = y + (size_t)k * D_DIM;
        float p = 0.f;
        #pragma unroll 4
        for (int i = 0; i < 64; ++i) {
            const int d = chunk * 64 + i;
            p += om[(size_t)d * S_DIM + s] * yk[d];
        }
        atomicAdd(&lds_v[s], p);
    }
    __syncthreads();
    if (t < S_DIM) v_ws[k * S_DIM + t] = lds_v[t];
}

// ---------------------------------------------------------------------------
// Kernel B: one block per (64-row x-tile, prototype k). 16 waves.
//   Double-buffered async staging of X (64x128) and Omega (128x64) chunks.
//   GEMM1: A = Xtile @ Omega_k,  u = v_k - A  (LDS)
//   GEMM2: W = u @ G_k
//   dist[b] = ||y_k - x_b||^2 + sum_s u[b,s] * (W[b,s] - 2*u[b,s])
// ---------------------------------------------------------------------------
__global__ __launch_bounds__(512)
void tangent_dist_kernel(const float* __restrict__ x,       // [B, D]
                         const float* __restrict__ y,       // [K, D]
                         const float* __restrict__ omegas,  // [K, D, S]
                         const float* __restrict__ G_ws,    // [K, 64, 64]
                         const float* __restrict__ v_ws,    // [K, 64]
                         float* __restrict__ out)           // [B, K]
{
    __shared__ float lds_x[2][TILE_M * XPAD];    // 2 x 33 KB
    __shared__ float lds_om[2][DCHUNK * S_DIM];  // 2 x 32 KB
    __shared__ float lds_u[TILE_M * UPAD];       // 16.6 KB
    __shared__ float lds_dist[TILE_M];

    const int t      = threadIdx.x;
    const int lane   = t & 31;
    const int wave   = t >> 5;
    const int mi     = wave >> 2;
    const int ni     = wave & 3;
    const int lmod   = lane & 15;
    const int lhalf  = lane >> 4;
    const int koff   = lhalf * 2;
    const int m_base = blockIdx.x * TILE_M;
    const int k      = blockIdx.y;

    const float* om = omegas + (size_t)k * D_DIM * S_DIM;
    const float* yk = y + (size_t)k * D_DIM;
    const int n = ni * 16 + lmod;

    // Issue one chunk of async global->LDS copies: 8 b128 ops per wave
    // (4 for X: 64x128 floats, 4 for Omega: 128x64 floats).
    auto issue_chunk = [&](int ch, int buf) {
        #pragma unroll
        for (int i = 0; i < 4; ++i) {
            const int slot = (wave * 4 + i) * 32 + lane;   // 0..2047, 16B each
            // X chunk: row-major 64 x 128
            const int xe  = slot * 4;
            const int xr  = xe >> 7;
            const int xc  = xe & 127;
            async_copy16((unsigned)(uintptr_t)&lds_x[buf][xr * XPAD + xc],
                         &x[(size_t)(m_base + xr) * D_DIM + ch * DCHUNK + xc]);
            // Omega chunk: row-major 128 x 64 (16 slots per row)
            const int orr = slot >> 4;
            const int oc  = (slot & 15) * 4;
            async_copy16((unsigned)(uintptr_t)&lds_om[buf][orr * S_DIM + oc],
                         &om[(size_t)(ch * DCHUNK + orr) * S_DIM + oc]);
        }
    };

    if (t < TILE_M) lds_dist[t] = 0.f;
    issue_chunk(0, 0);

    v8f acc = {};
    for (int ch = 0; ch < NCHUNK; ++ch) {
        const int buf = ch & 1;
        if (ch + 1 < NCHUNK) {
            issue_chunk(ch + 1, buf ^ 1);   // prefetch next chunk
            WAIT_ASYNC(8);                  // chunk ch landed; next stays in flight
        } else {
            WAIT_ASYNC(0);
        }
        __syncthreads();

        // Partial ||y - x||^2 : 8 sub-chunks of 16 per row.
        {
            const int r   = t & 63;
            const int sub = t >> 6;
            float s = 0.f;
            #pragma unroll
            for (int i = 0; i < 16; ++i) {
                const int d  = sub * 16 + i;
                const float dd = yk[ch * DCHUNK + d] - lds_x[buf][r * XPAD + d];
                s += dd * dd;
            }
            atomicAdd(&lds_dist[r], s);
        }

        // GEMM1 partial: Xtile_chunk @ Omega_chunk (32 wmma steps, all-LDS)
        #pragma unroll 4
        for (int kk = 0; kk < DCHUNK; kk += 4) {
            v2f a, b;
            a.x = lds_x[buf][(mi * 16 + lmod) * XPAD + kk + koff];
            a.y = lds_x[buf][(mi * 16 + lmod) * XPAD + kk + koff + 1];
            b.x = lds_om[buf][(kk + koff)     * S_DIM + n];
            b.y = lds_om[buf][(kk + koff + 1) * S_DIM + n];
            acc = __builtin_amdgcn_wmma_f32_16x16x4_f32(false, a, false, b,
                                                        (short)0, acc, false, false);
        }
        __syncthreads();  // all waves done with buf before it is overwritten
    }

    // u = v_k - A  -> LDS
    {
        const float vn = v_ws[k * S_DIM + n];
        #pragma unroll
        for (int j = 0; j < 8; ++j) {
            const int row = mi * 16 + j + lhalf * 8;
            lds_u[row * UPAD + n] = vn - acc[j];
        }
    }
    __syncthreads();

    // GEMM2: W = u @ G_k (16 wmma steps), fold into per-row distance.
    {
        const float* Gk = G_ws + (size_t)k * S_DIM * S_DIM;
        v8f acc2 = {};
        #pragma unroll
        for (int kk = 0; kk < S_DIM; kk += 4) {
            v2f a, b;
            a.x = lds_u[(mi * 16 + lmod) * UPAD + kk + koff];
            a.y = lds_u[(mi * 16 + lmod) * UPAD + kk + koff + 1];
            b.x = Gk[(kk + koff)     * S_DIM + n];
            b.y = Gk[(kk + koff + 1) * S_DIM + n];
            acc2 = __builtin_amdgcn_wmma_f32_16x16x4_f32(false, a, false, b,
                                                         (short)0, acc2, false, false);
        }
        #pragma unroll
        for (int j = 0; j < 8; ++j) {
            const int row = mi * 16 + j + lhalf * 8;
            const float uv = lds_u[row * UPAD + n];
            atomicAdd(&lds_dist[row], uv * (acc2[j] - 2.f * uv));
        }
    }
    __syncthreads();

    if (t < TILE_M)
        out[(size_t)(m_base + t) * K_PROTO + k] = lds_dist[t];
}

// ---------------------------------------------------------------------------
extern "C" void kernel_launch(void* const* d_in, const int* in_sizes, int n_in,
                              void* d_out, int out_size, void* d_ws, size_t ws_size,
                              hipStream_t stream) {
    const float* x      = (const float*)d_in[0];  // [1024, 512]
    const float* y      = (const float*)d_in[1];  // [62, 512]
    const float* omegas = (const float*)d_in[2];  // [62, 512, 64]
    float* out  = (float*)d_out;                  // [1024, 62]
    float* G_ws = (float*)d_ws;                               // K*64*64 floats
    float* v_ws = G_ws + (size_t)K_PROTO * S_DIM * S_DIM;     // K*64 floats

    proto_prep_kernel<<<K_PROTO, 512, 0, stream>>>(y, omegas, G_ws, v_ws);
    tangent_dist_kernel<<<dim3(B_TOT / TILE_M, K_PROTO), 512, 0, stream>>>(
        x, y, omegas, G_ws, v_ws, out);
}